// SplineActivation_36112085024874
// MI455X (gfx1250) — compile-verified
//
#include <hip/hip_runtime.h>

// SplineActivation: y[b,d,k] = deBoor3(x[b,d]; knots, sum_p coeffs[:,p]), k=0..9.
// Memory-bound: 16 MB in + 160 MB out => ~7.6 us floor @ 23.3 TB/s.
// Pipeline per 256-thread block (2048 elements):
//   1) TDM: TENSOR_LOAD_TO_LDS pulls the block's 8 KB x-tile into LDS (wave 0 issues,
//      s_wait_tensorcnt 0, workgroup barrier).  OOB elements read as zero (tail-safe).
//   2) each thread evaluates 8 splines (3-case cndmask de Boor, reciprocal denominators
//      precomputed by a tiny setup kernel -> no per-lane divides), writes values to LDS.
//   3) transposed broadcast store: for step s, lane tid stores the v4f at word
//      4*(s*256+tid) of the block's contiguous 80 KB output region -> every
//      global_store_b128 is a fully coalesced 512 B/wave NT stream store.

typedef float v4f  __attribute__((ext_vector_type(4)));
typedef unsigned int u32x4 __attribute__((ext_vector_type(4)));
typedef int i32x4 __attribute__((ext_vector_type(4)));
typedef int i32x8 __attribute__((ext_vector_type(8)));

#define BLOCK 256
#define EPT   8
#define EB    (BLOCK * EPT)   /* 2048 elements per block */
#define KREP  10              /* broadcast factor (num_knots) */

// Per-case table (16 floats per case, cases ell = 3,4,5):
//  [0..2]  t[e-2], t[e-1], t[e]
//  [3..8]  1/(t[e+1]-t[e-2]), 1/(t[e+2]-t[e-1]), 1/(t[e+3]-t[e]),
//          1/(t[e+1]-t[e-1]), 1/(t[e+2]-t[e]),   1/(t[e+1]-t[e])
//  [9..12] csum[e-3..e]   [13,14] t[4], t[5]
__global__ void spline_setup_kernel(const float* __restrict__ t,
                                    const float* __restrict__ c,
                                    int P, float* __restrict__ tab)
{
    if (threadIdx.x != 0 || blockIdx.x != 0) return;
    float cs[6];
    for (int j = 0; j < 6; ++j) {
        float s = 0.f;
        for (int p = 0; p < P; ++p) s += c[j * P + p];
        cs[j] = s;
    }
    for (int k = 0; k < 3; ++k) {
        const int e = 3 + k;
        float* T = tab + 16 * k;
        T[0] = t[e - 2]; T[1] = t[e - 1]; T[2] = t[e];
        T[3] = 1.0f / (t[e + 1] - t[e - 2]);
        T[4] = 1.0f / (t[e + 2] - t[e - 1]);
        T[5] = 1.0f / (t[e + 3] - t[e]);
        T[6] = 1.0f / (t[e + 1] - t[e - 1]);
        T[7] = 1.0f / (t[e + 2] - t[e]);
        T[8] = 1.0f / (t[e + 1] - t[e]);
        T[9]  = cs[e - 3]; T[10] = cs[e - 2];
        T[11] = cs[e - 1]; T[12] = cs[e];
        T[13] = t[4]; T[14] = t[5]; T[15] = 0.f;
    }
}

__device__ __forceinline__ float sel3(bool b4, bool b5, float a, float b, float c)
{
    float r = b5 ? c : b;   // v_cndmask
    return b4 ? r : a;      // v_cndmask
}

__global__ void __launch_bounds__(BLOCK)
spline_bcast_kernel(const float* __restrict__ x, float* __restrict__ out,
                    const float* __restrict__ tab, long long n)
{
    __shared__ __align__(16) float lds_v[EB + 4];  // spline values (+pad for e0+1 read)
    __shared__ __align__(16) float lds_x[EB];      // TDM destination tile

    // Uniform case tables -> scalar cache / SGPRs.
    float T0[13], T1[13], T2[13];
#pragma unroll
    for (int j = 0; j < 13; ++j) {
        T0[j] = tab[j];
        T1[j] = tab[16 + j];
        T2[j] = tab[32 + j];
    }
    const float t4 = tab[13], t5 = tab[14];

    const int tid = threadIdx.x;
    const long long eb = (long long)blockIdx.x * EB;
    long long rem = n - eb;
    const int tile = (rem < (long long)EB) ? (int)rem : EB;   // elements this block

    // ---- Stage 1: TDM 1-D tile load, global -> LDS (wave 0 only; EXEC-independent) ----
    if (tid < 32) {
        const unsigned lds_off = (unsigned)(uintptr_t)(void*)lds_x;  // low 32b of flat
        const unsigned long long ga = (unsigned long long)(uintptr_t)(x + eb);

        u32x4 g0;
        g0.x = 1u;                                   // count=1 (valid), load, user mode
        g0.y = lds_off;                              // lds_addr
        g0.z = (unsigned)ga;                         // global_addr[31:0]
        g0.w = (unsigned)((ga >> 32) & 0x1FFFFFFull) // global_addr[56:32]
             | (2u << 30);                           // type=2 ("image")

        i32x8 g1;
        g1[0] = 0x00020000;                          // wg_mask=0, data_size=2 (4 B)
        g1[1] = (int)(((unsigned)tile & 0xFFFFu) << 16);        // tensor_dim0[15:0]
        g1[2] = (int)((((unsigned)tile >> 16) & 0xFFFFu))       // tensor_dim0[31:16]
              | (1 << 16);                                      // tensor_dim1 = 1
        g1[3] = (int)((unsigned)EB << 16);           // tile_dim0 = 2048 (OOB reads -> 0)
        g1[4] = 0;                                   // tile_dim1 = tile_dim2 = 0 (1-D)
        g1[5] = EB;                                  // tensor_dim0_stride (benign, 1-D)
        g1[6] = 0;
        g1[7] = 0;

        const i32x4 z4 = (i32x4)0;
#if __clang_major__ >= 23
        const i32x8 z8 = (i32x8)0;
        __builtin_amdgcn_tensor_load_to_lds(g0, g1, z4, z4, z8, 0);
#else
        __builtin_amdgcn_tensor_load_to_lds(g0, g1, z4, z4, 0);
#endif
        __builtin_amdgcn_s_wait_tensorcnt(0);
    }
    __syncthreads();

    // ---- Stage 2: evaluate 8 splines/thread from LDS, write values to LDS ----
    auto eval = [&](float xv) -> float {
        const bool b4 = xv >= t4;
        const bool b5 = xv >= t5;
        const float L0 = sel3(b4, b5, T0[0],  T1[0],  T2[0]);
        const float L1 = sel3(b4, b5, T0[1],  T1[1],  T2[1]);
        const float L2 = sel3(b4, b5, T0[2],  T1[2],  T2[2]);
        const float I1 = sel3(b4, b5, T0[3],  T1[3],  T2[3]);
        const float I2 = sel3(b4, b5, T0[4],  T1[4],  T2[4]);
        const float I3 = sel3(b4, b5, T0[5],  T1[5],  T2[5]);
        const float I4 = sel3(b4, b5, T0[6],  T1[6],  T2[6]);
        const float I5 = sel3(b4, b5, T0[7],  T1[7],  T2[7]);
        const float I6 = sel3(b4, b5, T0[8],  T1[8],  T2[8]);
        float d0 = sel3(b4, b5, T0[9],  T1[9],  T2[9]);
        float d1 = sel3(b4, b5, T0[10], T1[10], T2[10]);
        float d2 = sel3(b4, b5, T0[11], T1[11], T2[11]);
        float d3 = sel3(b4, b5, T0[12], T1[12], T2[12]);
        const float xm0 = xv - L0, xm1 = xv - L1, xm2 = xv - L2;
        const float a1 = xm0 * I1, a2 = xm1 * I2, a3 = xm2 * I3;   // r=1
        const float b2 = xm1 * I4, b3 = xm2 * I5;                  // r=2
        const float c3 = xm2 * I6;                                 // r=3
        d3 = d2 + a3 * (d3 - d2);   // r=1, j=3
        d2 = d1 + a2 * (d2 - d1);   // r=1, j=2
        d1 = d0 + a1 * (d1 - d0);   // r=1, j=1
        d3 = d2 + b3 * (d3 - d2);   // r=2, j=3
        d2 = d1 + b2 * (d2 - d1);   // r=2, j=2
        d3 = d2 + c3 * (d3 - d2);   // r=3, j=3
        return d3;
    };

    const v4f* lx = (const v4f*)(lds_x + tid * EPT);
    const v4f xa = lx[0];
    const v4f xb = lx[1];
    v4f* lv = (v4f*)(lds_v + tid * EPT);
    lv[0] = (v4f){eval(xa.x), eval(xa.y), eval(xa.z), eval(xa.w)};
    lv[1] = (v4f){eval(xb.x), eval(xb.y), eval(xb.z), eval(xb.w)};
    __syncthreads();

    // ---- Stage 3: transposed, fully-coalesced NT broadcast stores ----
    // Block's output region: KREP*EB = 20480 contiguous words. Step s, lane tid
    // stores words [4m, 4m+4), m = s*BLOCK+tid. 4m mod 10 is even; only r==8
    // straddles two elements.
    const int tile_words = tile * KREP;
    float* const oblk = out + eb * KREP;
#pragma unroll
    for (int s = 0; s < (EPT * KREP) / 4; ++s) {       // 20 steps
        const int m  = s * BLOCK + tid;
        const int w0 = 4 * m;
        const unsigned e0 = (unsigned)w0 / 10u;        // strength-reduced by compiler
        const int r = w0 - 10 * (int)e0;
        const float f0 = lds_v[e0];
        const float f1 = lds_v[e0 + 1];
        const float hi = (r == 8) ? f1 : f0;
        const v4f q = (v4f){f0, f0, hi, hi};
        if (w0 + 4 <= tile_words) {
            __builtin_nontemporal_store(q, (v4f*)(oblk + w0));   // 512 B/wave, NT
        } else if (w0 < tile_words) {                            // ragged tail only
            for (int k = 0; k < tile_words - w0; ++k) oblk[w0 + k] = q[k];
        }
    }
}

extern "C" void kernel_launch(void* const* d_in, const int* in_sizes, int n_in,
                              void* d_out, int out_size, void* d_ws, size_t ws_size,
                              hipStream_t stream)
{
    (void)n_in; (void)out_size; (void)ws_size;

    const float* x      = (const float*)d_in[0];
    const float* knots  = (const float*)d_in[1];
    const float* coeffs = (const float*)d_in[2];
    float*       out    = (float*)d_out;
    float*       tab    = (float*)d_ws;    // 48 floats of scratch

    const int num_knots = in_sizes[1];                       // 10 here
    const int P = (num_knots > 0) ? (in_sizes[2] / num_knots) : 1;
    const long long n = (long long)in_sizes[0];

    spline_setup_kernel<<<1, 32, 0, stream>>>(knots, coeffs, P, tab);

    const long long grid = (n + EB - 1) / EB;                // 2048 blocks here
    spline_bcast_kernel<<<(int)grid, BLOCK, 0, stream>>>(x, out, tab, n);
}